// KANNeuron_21114059227631
// MI455X (gfx1250) — compile-verified
//
#include <hip/hip_runtime.h>
#include <stdint.h>

#define BATCH  16384
#define IN_DIM 512
#define NDEG   9                    // degree 8 -> 9 Chebyshev terms
#define NCOEF  (IN_DIM * NDEG)      // 4608 floats = 18 KB
#define BLOCK  256
#define WAVES  8                    // wave32 -> 8 waves per block
#define ROWS_PER_BLOCK 16
#define NBLOCKS (BATCH / ROWS_PER_BLOCK)   // 1024 blocks

typedef __attribute__((ext_vector_type(4))) unsigned int v4u;
typedef __attribute__((ext_vector_type(4))) int          v4i;
typedef __attribute__((ext_vector_type(8))) int          v8i;
typedef __attribute__((ext_vector_type(2))) float        f2;

// ---- VOP3P packed f32 math (2 lanes of Chebyshev per instruction) ----
__device__ __forceinline__ f2 pk_fma(f2 a, f2 b, f2 c) {     // a*b + c
    f2 d;
    asm("v_pk_fma_f32 %0, %1, %2, %3" : "=v"(d) : "v"(a), "v"(b), "v"(c));
    return d;
}
__device__ __forceinline__ f2 pk_fms(f2 a, f2 b, f2 c) {     // a*b - c (NEG on src2)
    f2 d;
    asm("v_pk_fma_f32 %0, %1, %2, %3 neg_lo:[0,0,1] neg_hi:[0,0,1]"
        : "=v"(d) : "v"(a), "v"(b), "v"(c));
    return d;
}
__device__ __forceinline__ f2 pk_add(f2 a, f2 b) {           // a + b
    f2 d;
    asm("v_pk_add_f32 %0, %1, %2" : "=v"(d) : "v"(a), "v"(b));
    return d;
}

__global__ __launch_bounds__(BLOCK) void cheb_kan_kernel(
    const float* __restrict__ x,
    const float* __restrict__ coef,
    float* __restrict__ out)
{
    __shared__ float stage[NCOEF];        // TDM target: linear copy of coef
    __shared__ f2    c_int[NCOEF / 2];    // pair-interleaved: {c[d,k], c[d+32,k]}

    // ---- Stage the 18KB coefficient table global->LDS with the Tensor Data
    // ---- Mover: one DMA instruction per block, tracked by TENSORcnt.
    if (threadIdx.x < 32) {  // first wave issues it (TDM ignores EXEC; D# in SGPRs)
        uint32_t lds_off = (uint32_t)(uintptr_t)(&stage[0]);       // flat addr[31:0] == LDS offset
        unsigned long long ga = (unsigned long long)(uintptr_t)coef;

        // D# group 0: count=1 (valid, user mode), lds_addr, global_addr[56:0], type=2
        v4u g0;
        g0.x = 1u;
        g0.y = lds_off;
        g0.z = (unsigned)(ga & 0xffffffffull);
        g0.w = (unsigned)((ga >> 32) & 0x1ffffffull) | (2u << 30);

        // D# group 1: data_size=4B; tensor_dim0 = tile_dim0 = 4608; dim1 = 1;
        //             tensor_dim0_stride = 4608; no multicast / barrier / pad / iterate
        v8i g1;
        g1[0] = (int)(2u << 16);                  // data_size code 2 = 4 bytes
        g1[1] = (int)((unsigned)NCOEF << 16);     // tensor_dim0[15:0] (bits 63:48)
        g1[2] = (int)(1u << 16);                  // tensor_dim1 = 1   (bits 111:80)
        g1[3] = (int)((unsigned)NCOEF << 16);     // tile_dim0         (bits 127:112)
        g1[4] = 0;                                // tile_dim1/2 unused
        g1[5] = NCOEF;                            // tensor_dim0_stride low 32
        g1[6] = 0;
        g1[7] = 0;

        v4i g2 = {0, 0, 0, 0};
        v4i g3 = {0, 0, 0, 0};
#if __clang_major__ >= 23
        v8i g4 = {0, 0, 0, 0, 0, 0, 0, 0};
        __builtin_amdgcn_tensor_load_to_lds(g0, g1, g2, g3, g4, 0);
#else
        __builtin_amdgcn_tensor_load_to_lds(g0, g1, g2, g3, 0);
#endif
        __builtin_amdgcn_s_wait_tensorcnt(0);     // s_wait_tensorcnt 0
    }
    __syncthreads();

    // ---- One-time LDS->LDS permute into pk-friendly interleave.
    // Pair p = jp*32 + l covers dims (jp*64 + l, jp*64 + l + 32); unit k holds
    // {c[d,k], c[d+32,k]} as one b64. Loads stride 9, stores stride 18: both
    // conflict-free across a wave32.
    {
        const int p   = threadIdx.x;                       // 256 pairs
        const int dlo = ((p >> 5) << 6) + (p & 31);        // jp*64 + l
#pragma unroll
        for (int k = 0; k < NDEG; ++k) {
            f2 v;
            v.x = stage[dlo * NDEG + k];
            v.y = stage[(dlo + 32) * NDEG + k];
            c_int[p * NDEG + k] = v;                       // ds_store_b64
        }
    }
    __syncthreads();

    const int lane = threadIdx.x & 31;
    const int wave = threadIdx.x >> 5;
    const int row0 = blockIdx.x * ROWS_PER_BLOCK + wave;

#pragma unroll
    for (int i = 0; i < ROWS_PER_BLOCK / WAVES; ++i) {
        const int row = row0 + i * WAVES;
        const float* __restrict__ xrow = x + (size_t)row * IN_DIM;

        // Warm the next row this wave will touch (global_prefetch_b8).
        if (row + WAVES < BATCH)
            __builtin_prefetch(xrow + (size_t)WAVES * IN_DIM + lane * 16, 0, 0);

        // Coalesced stream of the row: lane L covers dims L, L+32, ..., L+480.
        float xv[16];
#pragma unroll
        for (int j = 0; j < 16; ++j)
            xv[j] = xrow[lane + 32 * j];

        // Two independent Chebyshev chains per instruction via V_PK_FMA_F32;
        // every coefficient operand is one ds_load_b64 straight into the pair.
        f2 acc2 = {0.0f, 0.0f};
#pragma unroll
        for (int jp = 0; jp < 8; ++jp) {
            const int j0 = 2 * jp;                          // dims (lane+64jp, +32)
            const f2* __restrict__ cp = &c_int[(32 * jp + lane) * NDEG];

            f2 xj;  xj.x = xv[j0];  xj.y = xv[j0 + 1];
            f2 x2 = pk_add(xj, xj);
            f2 tm2 = {1.0f, 1.0f};
            f2 tm1 = xj;
            f2 s = pk_fma(cp[1], xj, cp[0]);      // c0*T0 + c1*T1
#pragma unroll
            for (int k = 2; k < NDEG; ++k) {
                f2 t = pk_fms(x2, tm1, tm2);      // T_k = 2x*T_{k-1} - T_{k-2}
                s = pk_fma(cp[k], t, s);
                tm2 = tm1;
                tm1 = t;
            }
            acc2 = pk_add(acc2, s);
        }
        float acc = acc2.x + acc2.y;

        // wave32 all-lane reduction of the per-lane partials
#pragma unroll
        for (int off = 16; off > 0; off >>= 1)
            acc += __shfl_xor(acc, off, 32);

        if (lane == 0)
            out[row] = acc;
    }
}

extern "C" void kernel_launch(void* const* d_in, const int* in_sizes, int n_in,
                              void* d_out, int out_size, void* d_ws, size_t ws_size,
                              hipStream_t stream) {
    (void)in_sizes; (void)n_in; (void)out_size; (void)d_ws; (void)ws_size;
    const float* x    = (const float*)d_in[0];   // (16384, 512) f32
    const float* coef = (const float*)d_in[1];   // (4608,) f32
    // d_in[2] is `degree` (== 8), baked into the kernel.
    float* out = (float*)d_out;                  // (16384,) f32
    cheb_kan_kernel<<<NBLOCKS, BLOCK, 0, stream>>>(x, coef, out);
}